// AttentionUnit_v2_44083544326379
// MI455X (gfx1250) — compile-verified
//
#include <hip/hip_runtime.h>
#include <hip/hip_bf16.h>

typedef __attribute__((ext_vector_type(2))) float v2f;
typedef __attribute__((ext_vector_type(4))) float v4f;
typedef __attribute__((ext_vector_type(8))) float v8f;

#define QUERY_DIM 512
#define KEY_DIM   256
#define N_BATCH   2048
#define N_KEYS    512
#define CHUNK     64
#define NCHUNK    (N_KEYS / CHUNK)   // 8

// ---------------------------------------------------------------------------
// Kernel A: tq = queries @ W.T + bias   (M=2048, K=512, N=256)
// One wave per 16x16 tile via V_WMMA_F32_16X16X4_F32 (unchanged from round 1:
// emits 4x v_wmma per unrolled loop body, clause-batched b64 loads).
// ---------------------------------------------------------------------------
__global__ __launch_bounds__(256) void linear_wmma_kernel(
    const float* __restrict__ q, const float* __restrict__ W,
    const float* __restrict__ bias, float* __restrict__ tq)
{
    const int wid   = blockIdx.x * 8 + (threadIdx.x >> 5);
    const int lane  = threadIdx.x & 31;
    const int tileM = wid >> 4;
    const int tileN = wid & 15;
    const int laneH = lane & 15;
    const int kOff  = (lane < 16) ? 0 : 2;

    const float* qrow = q + (size_t)(tileM * 16 + laneH) * QUERY_DIM + kOff;
    const float* wrow = W + (size_t)(tileN * 16 + laneH) * QUERY_DIM + kOff;

    v8f c = {};
#pragma unroll 4
    for (int k0 = 0; k0 < QUERY_DIM; k0 += 4) {
        v2f a, b;
        a.x = qrow[k0]; a.y = qrow[k0 + 1];
        b.x = wrow[k0]; b.y = wrow[k0 + 1];
        c = __builtin_amdgcn_wmma_f32_16x16x4_f32(
                false, a, false, b, (short)0, c, false, false);
    }

    const int   col     = tileN * 16 + laneH;
    const float bv      = bias[col];
    const int   rowBase = tileM * 16 + ((lane < 16) ? 0 : 8);
#pragma unroll
    for (int v = 0; v < 8; ++v)
        tq[(size_t)(rowBase + v) * KEY_DIM + col] = c[v] + bv;
}

// ---------------------------------------------------------------------------
// Async global->LDS helpers (CDNA5 ASYNCcnt path)
// ---------------------------------------------------------------------------
__device__ __forceinline__ void async_load_b128(unsigned lds_byte,
                                                const float* sbase,
                                                unsigned voff_byte) {
    // GVS mode: mem = SADDR + VADDR[31:0]; dest LDS byte addr in first VGPR.
    asm volatile("global_load_async_to_lds_b128 %0, %1, %2"
                 :: "v"(lds_byte), "v"(voff_byte), "s"(sbase)
                 : "memory");
}
__device__ __forceinline__ void wait_async_le(int tag) {
    if (tag) asm volatile("s_wait_asynccnt 16" ::: "memory");
    else     asm volatile("s_wait_asynccnt 0"  ::: "memory");
}

// ---------------------------------------------------------------------------
// Kernel B: single-pass fused attention. One 256-thread block per batch.
// keys streamed from HBM exactly ONCE into double-buffered LDS chunks via
// async loads; scores + online softmax + weighted sum all read from LDS.
// ---------------------------------------------------------------------------
__global__ __launch_bounds__(256) void attn_fused_kernel(
    const float* __restrict__ keys, const int* __restrict__ null_mask,
    const float* __restrict__ tq,
    float* __restrict__ attened, float* __restrict__ alpha_out)
{
    extern __shared__ float smem[];
    float* kbuf[2];
    kbuf[0]        = smem;                              // CHUNK*KEY_DIM
    kbuf[1]        = smem + CHUNK * KEY_DIM;            // CHUNK*KEY_DIM
    float* tq_s    = smem + 2 * CHUNK * KEY_DIM;        // KEY_DIM
    float* alpha_s = tq_s + KEY_DIM;                    // N_KEYS (raw scores)
    float* p_s     = alpha_s + N_KEYS;                  // CHUNK
    float* red     = p_s + CHUNK;                       // 256

    const int b    = blockIdx.x;
    const int tid  = threadIdx.x;
    const int lane = tid & 31;
    const int wave = tid >> 5;
    const float* kb = keys + (size_t)b * N_KEYS * KEY_DIM;
    const float NEG_INF = -__builtin_inff();

    tq_s[tid] = tq[(size_t)b * KEY_DIM + tid];

    // LDS byte base addresses of the two chunk buffers (low 32 bits of flat
    // shared pointer == wave-relative LDS offset).
    const unsigned kbufByte[2] = { (unsigned)(size_t)(void*)kbuf[0],
                                   (unsigned)(size_t)(void*)kbuf[1] };

    // issue async loads for one 64-row chunk: each wave owns 8 rows,
    // each lane moves 16B, 2 x b128 per row per wave.
    auto issue_chunk = [&](int cidx) {
        const unsigned lbase = kbufByte[cidx & 1];
#pragma unroll
        for (int rr = 0; rr < 8; ++rr) {
            const int row = wave * 8 + rr;                       // row in chunk
            const unsigned goff = (unsigned)((cidx * CHUNK + row) * (KEY_DIM * 4))
                                + (unsigned)(lane * 16);
            const unsigned loff = lbase + (unsigned)(row * (KEY_DIM * 4))
                                + (unsigned)(lane * 16);
            async_load_b128(loff,        kb, goff);
            async_load_b128(loff + 512u, kb, goff + 512u);
        }
    };

    issue_chunk(0);

    float m_run = NEG_INF;
    float s_run = 0.0f;
    float acc   = 0.0f;

    // per-lane slice of tq for the score dot products
    __syncthreads();   // tq_s visible
    const float* tql = tq_s + lane * 8;
    const float t0 = tql[0], t1 = tql[1], t2 = tql[2], t3 = tql[3];
    const float t4 = tql[4], t5 = tql[5], t6 = tql[6], t7 = tql[7];

    for (int c = 0; c < NCHUNK; ++c) {
        const int   cb    = c * CHUNK;
        float*      kcur  = kbuf[c & 1];
        const bool  more  = (c + 1 < NCHUNK);

        if (more) issue_chunk(c + 1);
        wait_async_le(more ? 1 : 0);   // chunk c resident (in-order completion)
        __syncthreads();

        // ---- scores for this chunk, read from LDS: 8 waves x 8 rows ----
#pragma unroll
        for (int rr = 0; rr < 8; ++rr) {
            const int r = wave * 8 + rr;
            const float* krow = kcur + r * KEY_DIM + lane * 8;
            v4f k0 = *(const v4f*)(krow);
            v4f k1 = *(const v4f*)(krow + 4);
            float s = k0.x * t0 + k0.y * t1 + k0.z * t2 + k0.w * t3
                    + k1.x * t4 + k1.y * t5 + k1.z * t6 + k1.w * t7;
            for (int off = 16; off > 0; off >>= 1)
                s += __shfl_xor(s, off, 32);
            if (lane == 0) {
                const int n = cb + r;
                alpha_s[n] = null_mask[(size_t)b * N_KEYS + n] ? NEG_INF : s;
            }
        }
        __syncthreads();

        // ---- chunk max (64 values) ----
        if (tid < CHUNK) red[tid] = alpha_s[cb + tid];
        __syncthreads();
        for (int s = 32; s > 0; s >>= 1) {
            if (tid < s) red[tid] = fmaxf(red[tid], red[tid + s]);
            __syncthreads();
        }
        const float cmax  = red[0];
        const float m_new = fmaxf(m_run, cmax);
        __syncthreads();

        if (m_new > NEG_INF) {    // uniform branch; fully-masked chunk -> skip
            const float f = __expf(m_run - m_new);   // exp(-inf)=0 on first hit
            if (tid < CHUNK) {
                const float pv = __expf(alpha_s[cb + tid] - m_new);
                p_s[tid] = pv;
                red[tid] = pv;
            }
            __syncthreads();
            for (int s = 32; s > 0; s >>= 1) {
                if (tid < s) red[tid] += red[tid + s];
                __syncthreads();
            }
            s_run = s_run * f + red[0];
            m_run = m_new;
            acc  *= f;
            // ---- weighted key sum from LDS: thread tid owns column tid ----
#pragma unroll 8
            for (int r = 0; r < CHUNK; ++r)
                acc = fmaf(p_s[r], kcur[r * KEY_DIM + tid], acc);
        }
        __syncthreads();   // all reads of kcur/p_s/red done before reuse
    }

    const float invs = 1.0f / s_run;
    attened[(size_t)b * KEY_DIM + tid] = acc * invs;
    alpha_out[(size_t)b * N_KEYS + tid]       = __expf(alpha_s[tid]       - m_run) * invs;
    alpha_out[(size_t)b * N_KEYS + tid + 256] = __expf(alpha_s[tid + 256] - m_run) * invs;
}

extern "C" void kernel_launch(void* const* d_in, const int* in_sizes, int n_in,
                              void* d_out, int out_size, void* d_ws, size_t ws_size,
                              hipStream_t stream) {
    const float* queries   = (const float*)d_in[0];        // (2048, 512)
    const float* keys      = (const float*)d_in[1];        // (2048, 512, 256)
    /* d_in[2] = trans_keys: unused by the reference */
    const int*   null_mask = (const int*)d_in[3];          // (2048, 512)
    const float* W         = (const float*)d_in[4];        // (256, 512)
    const float* bias      = (const float*)d_in[5];        // (256,)

    float* out      = (float*)d_out;
    float* attened  = out;                                 // (2048, 256)
    float* alpha_sm = out + (size_t)N_BATCH * KEY_DIM;     // (2048, 512)
    float* tq       = (float*)d_ws;                        // (2048, 256) scratch

    linear_wmma_kernel<<<256, 256, 0, stream>>>(queries, W, bias, tq);

    const size_t smemBytes = (size_t)(2 * CHUNK * KEY_DIM + KEY_DIM + N_KEYS
                                      + CHUNK + 256) * sizeof(float); // ~135 KB
    attn_fused_kernel<<<N_BATCH, 256, smemBytes, stream>>>(keys, null_mask, tq,
                                                           attened, alpha_sm);
}